// GATEncoder_80453327388953
// MI455X (gfx1250) — compile-verified
//
#include <hip/hip_runtime.h>
#include <hip/hip_bf16.h>
#include <math.h>

#define NN 100000
#define NE 1250000
#define DIM 64
#define NEG_SLOPE 0.2f
#define SM_EPS 1e-16f

typedef float v2f __attribute__((ext_vector_type(2)));
typedef float v8f __attribute__((ext_vector_type(8)));

// ---------------- float -> orderable uint key (for atomic max over floats) ---
__device__ __forceinline__ unsigned f32_key(float f) {
  unsigned u = __float_as_uint(f);
  return (u & 0x80000000u) ? ~u : (u | 0x80000000u);
}
__device__ __forceinline__ float key_f32(unsigned k) {
  unsigned u = (k & 0x80000000u) ? (k & 0x7fffffffu) : ~k;
  return __uint_as_float(u);
}

__device__ __forceinline__ void atomic_add_f32(float* p, float v) {
  __hip_atomic_fetch_add(p, v, __ATOMIC_RELAXED, __HIP_MEMORY_SCOPE_AGENT);
}
__device__ __forceinline__ void atomic_max_u32(unsigned* p, unsigned v) {
  __hip_atomic_fetch_max(p, v, __ATOMIC_RELAXED, __HIP_MEMORY_SCOPE_AGENT);
}

// ---------------- GEMM via WMMA f32 16x16x4:  H[n][o] = sum_k act(X[n][k]) * W[o][k]
// One wave per 16x16 tile. Block = 256 threads = 8 waves = 2 M-tiles x 4 N-tiles.
template <bool RELU>
__global__ __launch_bounds__(256) void gat_gemm_wmma(
    const float* __restrict__ X, const float* __restrict__ W,
    float* __restrict__ H) {
  const int lane  = threadIdx.x & 31;
  const int wave  = threadIdx.x >> 5;
  const int mTile = blockIdx.x * 2 + (wave >> 2);
  const int n0    = (wave & 3) * 16;
  const int m0    = mTile * 16;
  if (m0 >= NN) return;

  const int col  = lane & 15;   // M row (A) / N col (B,D)
  const int half = lane >> 4;   // selects K pair {0,1} vs {2,3} (A/B), M+8 (D)

  const float* arow = X + (size_t)(m0 + col) * DIM + 2 * half;  // x row-major
  const float* brow = W + (size_t)(n0 + col) * DIM + 2 * half;  // B[k][n]=W[n][k]

  v8f c = {};
#pragma unroll
  for (int k0 = 0; k0 < DIM; k0 += 4) {
    v2f a, b;
    a.x = arow[k0 + 0];
    a.y = arow[k0 + 1];
    if (RELU) { a.x = fmaxf(a.x, 0.f); a.y = fmaxf(a.y, 0.f); }
    b.x = brow[k0 + 0];
    b.y = brow[k0 + 1];
    c = __builtin_amdgcn_wmma_f32_16x16x4_f32(false, a, false, b,
                                              (short)0, c, false, false);
  }
  // D layout: VGPR r -> row m0 + r + 8*half, col n0 + (lane&15)
  float* hout = H + (size_t)m0 * DIM + n0 + col;
#pragma unroll
  for (int r = 0; r < 8; ++r) hout[(size_t)(r + 8 * half) * DIM] = c[r];
}

// ---------------- per-node attention scores: s = h . a  ----------------------
__global__ __launch_bounds__(256) void gat_scores(
    const float* __restrict__ H, const float* __restrict__ a_src,
    const float* __restrict__ a_dst, float* __restrict__ s_src,
    float* __restrict__ s_dst) {
  int n = blockIdx.x * blockDim.x + threadIdx.x;
  if (n >= NN) return;
  const float4* hp = (const float4*)(H + (size_t)n * DIM);
  const float4* as = (const float4*)a_src;
  const float4* ad = (const float4*)a_dst;
  float ss = 0.f, sd = 0.f;
#pragma unroll
  for (int q = 0; q < DIM / 4; ++q) {
    float4 h4 = hp[q], s4 = as[q], d4 = ad[q];
    ss += h4.x * s4.x + h4.y * s4.y + h4.z * s4.z + h4.w * s4.w;
    sd += h4.x * d4.x + h4.y * d4.y + h4.z * d4.z + h4.w * d4.w;
  }
  s_src[n] = ss;
  s_dst[n] = sd;
}

// ---------------- per-edge leaky-relu score + segment max --------------------
__global__ __launch_bounds__(256) void gat_edge_max(
    const int* __restrict__ ei, const float* __restrict__ s_src,
    const float* __restrict__ s_dst, unsigned* __restrict__ seg_max,
    float* __restrict__ ebuf) {
  int e = blockIdx.x * blockDim.x + threadIdx.x;
  if (e >= NE) return;
  int s = ei[e], d = ei[NE + e];
  float raw = s_src[s] + s_dst[d];
  float lr  = raw >= 0.f ? raw : NEG_SLOPE * raw;
  ebuf[e] = lr;
  atomic_max_u32(seg_max + d, f32_key(lr));
}

// ---------------- per-edge exp(score - max) + segment sum --------------------
__global__ __launch_bounds__(256) void gat_edge_exp(
    const int* __restrict__ ei, const unsigned* __restrict__ seg_max,
    float* __restrict__ ebuf, float* __restrict__ seg_sum) {
  int e = blockIdx.x * blockDim.x + threadIdx.x;
  if (e >= NE) return;
  int d = ei[NE + e];
  float v = expf(ebuf[e] - key_f32(seg_max[d]));
  ebuf[e] = v;
  atomic_add_f32(seg_sum + d, v);
}

// ---------------- finalize per-edge coefficient ------------------------------
__global__ __launch_bounds__(256) void gat_edge_coef(
    const int* __restrict__ ei, const float* __restrict__ seg_sum,
    const float* __restrict__ edge_attr, float* __restrict__ ebuf) {
  int e = blockIdx.x * blockDim.x + threadIdx.x;
  if (e >= NE) return;
  int d = ei[NE + e];
  ebuf[e] = ebuf[e] / (seg_sum[d] + SM_EPS) * edge_attr[e];
}

// ---------------- scatter: out[dst] += coef * h[src] (L2-resident atomics) ---
__global__ __launch_bounds__(256) void gat_scatter(
    const int* __restrict__ ei, const float* __restrict__ ebuf,
    const float* __restrict__ H, float* __restrict__ out) {
  int gid = blockIdx.x * blockDim.x + threadIdx.x;   // NE * 16 threads
  if (gid >= NE * 16) return;
  int e = gid >> 4, q = gid & 15;
  int s = ei[e], d = ei[NE + e];
  float coef = ebuf[e];
  float4 h4 = ((const float4*)(H + (size_t)s * DIM))[q];
  float* o = out + (size_t)d * DIM + q * 4;
  atomic_add_f32(o + 0, h4.x * coef);
  atomic_add_f32(o + 1, h4.y * coef);
  atomic_add_f32(o + 2, h4.z * coef);
  atomic_add_f32(o + 3, h4.w * coef);
}

extern "C" void kernel_launch(void* const* d_in, const int* in_sizes, int n_in,
                              void* d_out, int out_size, void* d_ws,
                              size_t ws_size, hipStream_t stream) {
  const float* x      = (const float*)d_in[0];
  const int*   ei     = (const int*)d_in[1];
  const float* eattr  = (const float*)d_in[2];
  const float* W1     = (const float*)d_in[3];
  const float* a_src1 = (const float*)d_in[4];
  const float* a_dst1 = (const float*)d_in[5];
  const float* W2     = (const float*)d_in[6];
  const float* a_src2 = (const float*)d_in[7];
  const float* a_dst2 = (const float*)d_in[8];

  // workspace layout
  float*    h       = (float*)d_ws;              // NN*DIM
  float*    agg1    = h + (size_t)NN * DIM;      // NN*DIM
  float*    s_src   = agg1 + (size_t)NN * DIM;   // NN
  float*    s_dst   = s_src + NN;                // NN
  float*    seg_sum = s_dst + NN;                // NN
  unsigned* seg_max = (unsigned*)(seg_sum + NN); // NN
  float*    ebuf    = (float*)(seg_max + NN);    // NE

  const int NB_NODE = (NN + 255) / 256;
  const int NB_EDGE = (NE + 255) / 256;
  const int NB_SCAT = (NE * 16 + 255) / 256;
  const int NB_GEMM = (NN / 16 + 1) / 2;         // 2 M-tiles per block

  auto run_layer = [&](const float* xin, const float* W, const float* as,
                       const float* ad, float* accum, bool relu_in) {
    hipMemsetAsync(seg_max, 0, (size_t)NN * 4, stream);
    hipMemsetAsync(seg_sum, 0, (size_t)NN * 4, stream);
    hipMemsetAsync(accum, 0, (size_t)NN * DIM * 4, stream);
    if (relu_in)
      gat_gemm_wmma<true><<<NB_GEMM, 256, 0, stream>>>(xin, W, h);
    else
      gat_gemm_wmma<false><<<NB_GEMM, 256, 0, stream>>>(xin, W, h);
    gat_scores<<<NB_NODE, 256, 0, stream>>>(h, as, ad, s_src, s_dst);
    gat_edge_max<<<NB_EDGE, 256, 0, stream>>>(ei, s_src, s_dst, seg_max, ebuf);
    gat_edge_exp<<<NB_EDGE, 256, 0, stream>>>(ei, seg_max, ebuf, seg_sum);
    gat_edge_coef<<<NB_EDGE, 256, 0, stream>>>(ei, seg_sum, eattr, ebuf);
    gat_scatter<<<NB_SCAT, 256, 0, stream>>>(ei, ebuf, h, accum);
  };

  run_layer(x, W1, a_src1, a_dst1, agg1, /*relu_in=*/false);
  run_layer(agg1, W2, a_src2, a_dst2, (float*)d_out, /*relu_in=*/true);
}